// GCNLayer_44032004718671
// MI455X (gfx1250) — compile-verified
//
#include <hip/hip_runtime.h>

#define N_NODES 20000
#define DEGREE  32
#define F       128   // in feats == out feats

typedef float v2f __attribute__((ext_vector_type(2)));
typedef float v8f __attribute__((ext_vector_type(8)));

// ---------------------------------------------------------------------------
// Kernel 1: gated aggregation.
// One wave32 per node; lane owns 4 contiguous features.
//   logits[n,d] = dot(h[nbr], W_gate[n]) + b_gate[n]
//   mask = round(sigmoid(logit)) == (logit > 0)
//   h1[n] = sum_d mask * h[nbr] * norm[n]
// ---------------------------------------------------------------------------
__global__ __launch_bounds__(256) void gcn_gate_aggregate(
    const float* __restrict__ h,
    const int*   __restrict__ neighbors,
    const float* __restrict__ norm,
    const float* __restrict__ Wg,
    const float* __restrict__ bg,
    float*       __restrict__ h1)
{
    const int wave = threadIdx.x >> 5;
    const int lane = threadIdx.x & 31;
    const int node = blockIdx.x * 8 + wave;
    if (node >= N_NODES) return;

    const float4 wg  = *(const float4*)(Wg + (size_t)node * F + lane * 4);
    const float  bgn = bg[node];
    const float  nn  = norm[node];
    const int*   nb  = neighbors + (size_t)node * DEGREE;

    float4 acc = {0.f, 0.f, 0.f, 0.f};
    #pragma unroll 4
    for (int d = 0; d < DEGREE; ++d) {
        const int j = nb[d];
        const float4 m = *(const float4*)(h + (size_t)j * F + lane * 4);
        float p = m.x * wg.x + m.y * wg.y + m.z * wg.z + m.w * wg.w;
        // wave32 butterfly reduction: every lane ends with the full dot
        p += __shfl_xor(p, 16, 32);
        p += __shfl_xor(p,  8, 32);
        p += __shfl_xor(p,  4, 32);
        p += __shfl_xor(p,  2, 32);
        p += __shfl_xor(p,  1, 32);
        if (p + bgn > 0.0f) {
            acc.x += m.x; acc.y += m.y; acc.z += m.z; acc.w += m.w;
        }
    }
    float4 o = {acc.x * nn, acc.y * nn, acc.z * nn, acc.w * nn};
    *(float4*)(h1 + (size_t)node * F + lane * 4) = o;
}

// ---------------------------------------------------------------------------
// Kernel 2: h2 = h1 @ weight via V_WMMA_F32_16X16X4_F32.
// Block = 256 threads = 8 waves; wave w computes the 16x16 tile at
// (rows blockIdx.x*16, cols w*16). weight staged in LDS (64 KB).
// A-frag layout (ISA 7.12.2, 32-bit A 16x4): lane = row(lane%16), khalf(lane/16);
//   a.x = A[row][k + 2*khalf], a.y = A[row][k + 2*khalf + 1]
// B-frag (4x16, symmetric): b.x = B[k + 2*khalf][col], b.y = next K row.
// C/D: vgpr i -> M = i + 8*khalf, N = lane%16.
// ---------------------------------------------------------------------------
__global__ __launch_bounds__(256) void gcn_gemm_wmma(
    const float* __restrict__ h1,
    const float* __restrict__ W,
    float*       __restrict__ h2)
{
    __shared__ float Bs[F * F];   // 64 KB (within 320 KB WGP LDS)
    for (int i = threadIdx.x * 4; i < F * F; i += 256 * 4)
        *(float4*)(Bs + i) = *(const float4*)(W + i);
    __syncthreads();

    const int wave  = threadIdx.x >> 5;   // output column tile, 0..7
    const int lane  = threadIdx.x & 31;
    const int row   = lane & 15;
    const int khalf = lane >> 4;          // 0 or 1
    const int tm    = blockIdx.x;         // 0..1249 (20000 = 1250*16 exactly)

    const float* Arow = h1 + ((size_t)tm * 16 + row) * F + khalf * 2;
    const float* Bcol = Bs + wave * 16 + row;   // row == col index here

    v8f c = {};
    #pragma unroll
    for (int k = 0; k < F; k += 4) {
        v2f a, b;
        a.x = Arow[k];
        a.y = Arow[k + 1];
        b.x = Bcol[(k + khalf * 2)     * F];
        b.y = Bcol[(k + khalf * 2 + 1) * F];
        c = __builtin_amdgcn_wmma_f32_16x16x4_f32(
                /*neg_a=*/false, a, /*neg_b=*/false, b,
                /*c_mod=*/(short)0, c, /*reuse_a=*/false, /*reuse_b=*/false);
    }

    float* out = h2 + (size_t)tm * 16 * F + wave * 16;
    #pragma unroll
    for (int i = 0; i < 8; ++i)
        out[(size_t)(i + khalf * 8) * F + row] = c[i];
}

// ---------------------------------------------------------------------------
// Kernel 3: second gather-sum, * norm, + bias, ReLU.
// ---------------------------------------------------------------------------
__global__ __launch_bounds__(256) void gcn_gather_bias_relu(
    const float* __restrict__ h2,
    const int*   __restrict__ neighbors,
    const float* __restrict__ norm,
    const float* __restrict__ bias,
    float*       __restrict__ out)
{
    const int wave = threadIdx.x >> 5;
    const int lane = threadIdx.x & 31;
    const int node = blockIdx.x * 8 + wave;
    if (node >= N_NODES) return;

    const int* nb = neighbors + (size_t)node * DEGREE;
    float4 acc = {0.f, 0.f, 0.f, 0.f};
    #pragma unroll 4
    for (int d = 0; d < DEGREE; ++d) {
        const int j = nb[d];
        const float4 m = *(const float4*)(h2 + (size_t)j * F + lane * 4);
        acc.x += m.x; acc.y += m.y; acc.z += m.z; acc.w += m.w;
    }
    const float  nn = norm[node];
    const float4 b  = *(const float4*)(bias + lane * 4);
    float4 o;
    o.x = fmaxf(acc.x * nn + b.x, 0.f);
    o.y = fmaxf(acc.y * nn + b.y, 0.f);
    o.z = fmaxf(acc.z * nn + b.z, 0.f);
    o.w = fmaxf(acc.w * nn + b.w, 0.f);
    *(float4*)(out + (size_t)node * F + lane * 4) = o;
}

// ---------------------------------------------------------------------------
// Host launch: inputs in setup_inputs() order:
//   0: h [N,F] f32, 1: neighbors [N,D] i32, 2: norm [N,1] f32,
//   3: W_gate [N,F] f32, 4: b_gate [N] f32, 5: weight [F,F] f32, 6: bias [F] f32
// d_out: [N,F] f32. d_ws: h1 then h2 (10.24 MB each).
// ---------------------------------------------------------------------------
extern "C" void kernel_launch(void* const* d_in, const int* in_sizes, int n_in,
                              void* d_out, int out_size, void* d_ws, size_t ws_size,
                              hipStream_t stream) {
    const float* h    = (const float*)d_in[0];
    const int*   nbrs = (const int*)  d_in[1];
    const float* norm = (const float*)d_in[2];
    const float* Wg   = (const float*)d_in[3];
    const float* bg   = (const float*)d_in[4];
    const float* W    = (const float*)d_in[5];
    const float* bias = (const float*)d_in[6];
    float*       out  = (float*)d_out;

    const size_t feat_bytes = (size_t)N_NODES * F * sizeof(float);
    float* h1 = (float*)d_ws;
    float* h2 = (float*)((char*)d_ws + feat_bytes);

    const int nodes_per_block = 8;                       // 8 waves of 32
    const int nblocks = (N_NODES + nodes_per_block - 1) / nodes_per_block;

    gcn_gate_aggregate<<<nblocks, 256, 0, stream>>>(h, nbrs, norm, Wg, bg, h1);
    gcn_gemm_wmma<<<N_NODES / 16, 256, 0, stream>>>(h1, W, h2);
    gcn_gather_bias_relu<<<nblocks, 256, 0, stream>>>(h2, nbrs, norm, bias, out);
}